// MAGIC_34316788695619
// MI455X (gfx1250) — compile-verified
//
#include <hip/hip_runtime.h>
#include <hip/hip_bf16.h>
#include <math.h>

// ---------------------------------------------------------------------------
// Problem constants
// ---------------------------------------------------------------------------
#define NAG   4096
#define OBS   256
#define HID   256
#define GDIM1 64
#define H1    4
#define ACT   64
#define QMAXV 255.0f

typedef float v2f __attribute__((ext_vector_type(2)));
typedef float v8f __attribute__((ext_vector_type(8)));

__device__ __forceinline__ float sigm(float x) { return 1.0f / (1.0f + __expf(-x)); }
__device__ __forceinline__ float lrelu(float x) { return x > 0.0f ? x : 0.2f * x; }

// ---------------------------------------------------------------------------
// Generic fp32 WMMA GEMM: C[M,N] = A[M,K] @ B(, optionally B^T) + bias (+ C)
// One wave32 per 16x16 output tile, K stepped by 4 (V_WMMA_F32_16X16X4_F32).
// Flags are template params so the inner loop is straight-line (no per-iter
// scalar branches). K-loop unrolled x4 => 4 chained WMMAs (same acc) per iteration.
// A frag (ISA 7.12.2): lanes 0-15 hold K=k,k+1 ; lanes 16-31 hold K=k+2,k+3.
// B frag: VGPR0 = rows k (lanes 0-15) / k+2 (lanes 16-31); VGPR1 = k+1 / k+3.
// C frag: VGPR v -> row v (lanes 0-15) / v+8 (lanes 16-31), col = lane%16.
// ---------------------------------------------------------------------------
template <bool TRANSB, bool HAS_BIAS, bool ACC>
__global__ void gemm16_f32(const float* __restrict__ A, const float* __restrict__ B,
                           const float* __restrict__ bias, float* __restrict__ C,
                           int K, int lda, int ldb, int ldc) {
  const int lane = threadIdx.x & 31;
  const int half = lane >> 4;
  const int l16  = lane & 15;
  const int row0 = blockIdx.y << 4;
  const int col0 = blockIdx.x << 4;

  v8f acc = {};
  const float* Arow = A + (size_t)(row0 + l16) * lda + (half << 1);
  const float* Bp;
  if (TRANSB) {
    Bp = B + (size_t)(col0 + l16) * ldb + (half << 1);   // Bt[c][k]
  } else {
    Bp = B + (size_t)(half << 1) * ldb + (col0 + l16);   // B[k][c]
  }

#pragma unroll 4
  for (int k = 0; k < K; k += 4) {
    v2f a;
    a.x = Arow[k];
    a.y = Arow[k + 1];
    v2f b;
    if (TRANSB) {
      b.x = Bp[k];
      b.y = Bp[k + 1];
    } else {
      b.x = Bp[(size_t)k * ldb];
      b.y = Bp[(size_t)(k + 1) * ldb];
    }
    acc = __builtin_amdgcn_wmma_f32_16x16x4_f32(false, a, false, b,
                                                (short)0, acc, false, false);
  }
#pragma unroll
  for (int v = 0; v < 8; ++v) {
    const int r = row0 + v + (half << 3);
    const int c = col0 + l16;
    float val = acc[v];
    if (HAS_BIAS) val += bias[c];
    if (ACC) val += C[(size_t)r * ldc + c];
    C[(size_t)r * ldc + c] = val;
  }
}

// ---------------------------------------------------------------------------
// LSTM pointwise: gates [N,1024] -> h,c ; h also into concat[:, :256] & comm0
// ---------------------------------------------------------------------------
__global__ void lstm_pointwise(const float* __restrict__ gates, const float* __restrict__ c0,
                               float* __restrict__ h_out, float* __restrict__ c_out,
                               float* __restrict__ hc, float* __restrict__ comm0) {
  const int idx = blockIdx.x * 256 + threadIdx.x;
  const int n = idx >> 8;
  const int j = idx & 255;
  const float* g = gates + (size_t)n * 1024;
  const float ig = sigm(g[j]);
  const float fg = sigm(g[256 + j]);
  const float gg = tanhf(g[512 + j]);
  const float og = sigm(g[768 + j]);
  const float c2 = fg * c0[idx] + ig * gg;
  const float h2 = og * tanhf(c2);
  h_out[idx] = h2;
  c_out[idx] = c2;
  hc[(size_t)n * 512 + j] = h2;
  comm0[idx] = h2;
}

// ---------------------------------------------------------------------------
// Two-stage global min/max -> (scale, zero_point)
// ---------------------------------------------------------------------------
__global__ void minmax_stage1(const float* __restrict__ src, float* __restrict__ pmin,
                              float* __restrict__ pmax, int len) {
  __shared__ float smn[256], smx[256];
  float mn = 3.4e38f, mx = -3.4e38f;
  for (int i = blockIdx.x * blockDim.x + threadIdx.x; i < len; i += gridDim.x * blockDim.x) {
    const float v = src[i];
    mn = fminf(mn, v);
    mx = fmaxf(mx, v);
  }
  smn[threadIdx.x] = mn; smx[threadIdx.x] = mx;
  __syncthreads();
  for (int s = 128; s > 0; s >>= 1) {
    if (threadIdx.x < s) {
      smn[threadIdx.x] = fminf(smn[threadIdx.x], smn[threadIdx.x + s]);
      smx[threadIdx.x] = fmaxf(smx[threadIdx.x], smx[threadIdx.x + s]);
    }
    __syncthreads();
  }
  if (threadIdx.x == 0) { pmin[blockIdx.x] = smn[0]; pmax[blockIdx.x] = smx[0]; }
}

__global__ void minmax_stage2(const float* __restrict__ pmin, const float* __restrict__ pmax,
                              float* __restrict__ sc, int nb) {
  __shared__ float smn[256], smx[256];
  float mn = 3.4e38f, mx = -3.4e38f;
  for (int i = threadIdx.x; i < nb; i += 256) {
    mn = fminf(mn, pmin[i]);
    mx = fmaxf(mx, pmax[i]);
  }
  smn[threadIdx.x] = mn; smx[threadIdx.x] = mx;
  __syncthreads();
  for (int s = 128; s > 0; s >>= 1) {
    if (threadIdx.x < s) {
      smn[threadIdx.x] = fminf(smn[threadIdx.x], smn[threadIdx.x + s]);
      smx[threadIdx.x] = fmaxf(smx[threadIdx.x], smx[threadIdx.x + s]);
    }
    __syncthreads();
  }
  if (threadIdx.x == 0) {
    float lo = smn[0], hi = smx[0];
    if (lo == hi) { lo -= 0.01f; hi += 0.01f; }
    const float scale = (hi - lo) / QMAXV;
    sc[0] = scale;
    sc[1] = rintf(-lo / scale);
  }
}

// Dequantize + per-block partial qloss sums
__global__ void quant_apply(const float* __restrict__ src, float* __restrict__ dst,
                            const float* __restrict__ sc, float* __restrict__ losspart,
                            int len) {
  __shared__ float sl[256];
  const float scale = sc[0];
  const float zp = sc[1];
  float acc = 0.0f;
  for (int i = blockIdx.x * blockDim.x + threadIdx.x; i < len; i += gridDim.x * blockDim.x) {
    const float t = src[i];
    float q = rintf(t / scale + zp);
    q = fminf(fmaxf(q, 0.0f), QMAXV);
    const float d = (q - zp) * scale;
    dst[i] = d;
    acc += log2f(2.0f * QMAXV * fabsf(d) + 1.0f);
  }
  sl[threadIdx.x] = acc;
  __syncthreads();
  for (int s = 128; s > 0; s >>= 1) {
    if (threadIdx.x < s) sl[threadIdx.x] += sl[threadIdx.x + s];
    __syncthreads();
  }
  if (threadIdx.x == 0) losspart[blockIdx.x] = sl[0];
}

__global__ void loss_final(const float* __restrict__ a, const float* __restrict__ b,
                           float* __restrict__ out, int nb) {
  __shared__ float sl[256];
  float acc = 0.0f;
  for (int i = threadIdx.x; i < nb; i += 256) acc += a[i] + b[i];
  sl[threadIdx.x] = acc;
  __syncthreads();
  for (int s = 128; s > 0; s >>= 1) {
    if (threadIdx.x < s) sl[threadIdx.x] += sl[threadIdx.x + s];
    __syncthreads();
  }
  if (threadIdx.x == 0) out[0] = sl[0];
}

// ---------------------------------------------------------------------------
// GAT attention projections: ci[h,n] = sum_g hW[n,h*G+g]*ai[h,g]  (cj same)
// ---------------------------------------------------------------------------
__global__ void cicj_kernel(const float* __restrict__ hW, const float* __restrict__ ai,
                            const float* __restrict__ aj, float* __restrict__ ci,
                            float* __restrict__ cj, int n, int H, int G) {
  const int idx = blockIdx.x * blockDim.x + threadIdx.x;
  if (idx >= H * n) return;
  const int h = idx / n;
  const int node = idx - h * n;
  const float* hp = hW + (size_t)node * (H * G) + h * G;
  const float* aih = ai + h * G;
  const float* ajh = aj + h * G;
  float si = 0.0f, sj = 0.0f;
  for (int g = 0; g < G; ++g) {
    const float v = hp[g];
    si += v * aih[g];
    sj += v * ajh[g];
  }
  ci[h * n + node] = si;
  cj[h * n + node] = sj;
}

__global__ void colmax_kernel(const float* __restrict__ cj, float* __restrict__ cjmax, int n) {
  __shared__ float sm[256];
  const float* c = cj + blockIdx.x * n;
  float m = -3.4e38f;
  for (int i = threadIdx.x; i < n; i += 256) m = fmaxf(m, c[i]);
  sm[threadIdx.x] = m;
  __syncthreads();
  for (int s = 128; s > 0; s >>= 1) {
    if (threadIdx.x < s) sm[threadIdx.x] = fmaxf(sm[threadIdx.x], sm[threadIdx.x + s]);
    __syncthreads();
  }
  if (threadIdx.x == 0) cjmax[blockIdx.x] = sm[0];
}

// ---------------------------------------------------------------------------
// Fused softmax-attention * hW via WMMA (flash-style, complete graph):
//   out[i, cb+0:64] = (1/s_i) * sum_j exp(lrelu(ci_i+cj_j) - m_i) * hW[j, cb..]
// m_i = lrelu(ci_i + max_j cj_j) (leaky-relu is monotone). One wave per
// (16-row tile, head, 64-col chunk); prob fragment is the WMMA A operand.
// ---------------------------------------------------------------------------
template <int EACT>
__global__ void gat_attn(const float* __restrict__ hW, const float* __restrict__ ci,
                         const float* __restrict__ cj, const float* __restrict__ cjmax,
                         const float* __restrict__ bias, float* __restrict__ out,
                         int n, int G, int ldh, int ldc) {
  const int lane = threadIdx.x & 31;
  const int half = lane >> 4;
  const int l16  = lane & 15;
  const int i0   = blockIdx.x << 4;
  const int head = blockIdx.y;
  const int colbase = head * G + (blockIdx.z << 6);

  const float* cjh = cj + head * n;
  const float ci_l = ci[head * n + i0 + l16];
  const float mi = lrelu(ci_l + cjmax[head]);

  v8f acc[4] = {v8f{}, v8f{}, v8f{}, v8f{}};
  float rs = 0.0f;

#pragma unroll 2
  for (int jb = 0; jb < n; jb += 4) {
    const int j0 = jb + (half << 1);
    const float p0 = __expf(lrelu(ci_l + cjh[j0]) - mi);
    const float p1 = __expf(lrelu(ci_l + cjh[j0 + 1]) - mi);
    rs += p0 + p1;
    v2f a; a.x = p0; a.y = p1;
    const float* B0 = hW + (size_t)j0 * ldh + colbase + l16;
    const float* B1 = B0 + ldh;
#pragma unroll
    for (int t = 0; t < 4; ++t) {
      v2f b; b.x = B0[t * 16]; b.y = B1[t * 16];
      acc[t] = __builtin_amdgcn_wmma_f32_16x16x4_f32(false, a, false, b,
                                                     (short)0, acc[t], false, false);
    }
  }

  rs += __shfl_xor(rs, 16, 32);  // combine the two half-wave K partials per row
  __shared__ float sden[16];
  if (lane < 16) sden[lane] = rs;
  __syncthreads();

#pragma unroll
  for (int t = 0; t < 4; ++t) {
#pragma unroll
    for (int v = 0; v < 8; ++v) {
      const int r = v + (half << 3);
      const int row = i0 + r;
      const int ccol = colbase + t * 16 + l16;
      float val = acc[t][v] / sden[r] + bias[ccol];
      if (EACT) val = val > 0.0f ? val : expm1f(val);  // ELU(alpha=1)
      out[(size_t)row * ldc + ccol] = val;
    }
  }
}

// ---------------------------------------------------------------------------
// value[n] = concat(h,comm)[n,:] . Wv + bv
// ---------------------------------------------------------------------------
__global__ void value_kernel(const float* __restrict__ hc, const float* __restrict__ Wv,
                             const float* __restrict__ bv, float* __restrict__ val) {
  const int nrow = blockIdx.x * blockDim.x + threadIdx.x;
  if (nrow >= NAG) return;
  const float* r = hc + (size_t)nrow * 512;
  float s = 0.0f;
#pragma unroll 8
  for (int k = 0; k < 512; ++k) s += r[k] * Wv[k];
  val[nrow] = s + bv[0];
}

// ---------------------------------------------------------------------------
// Row-wise log-softmax over 64 logits, one wave per row (2 cols per lane)
// ---------------------------------------------------------------------------
__global__ void logsoftmax_kernel(const float* __restrict__ logits, float* __restrict__ out) {
  const int nrow = blockIdx.x;
  const int lane = threadIdx.x;
  const float x0 = logits[nrow * 64 + lane];
  const float x1 = logits[nrow * 64 + lane + 32];
  float m = fmaxf(x0, x1);
  for (int s = 16; s > 0; s >>= 1) m = fmaxf(m, __shfl_xor(m, s, 32));
  float sum = __expf(x0 - m) + __expf(x1 - m);
  for (int s = 16; s > 0; s >>= 1) sum += __shfl_xor(sum, s, 32);
  const float l = logf(sum);
  out[nrow * 64 + lane] = x0 - m - l;
  out[nrow * 64 + lane + 32] = x1 - m - l;
}

// ---------------------------------------------------------------------------
// Host-side orchestration
// ---------------------------------------------------------------------------
extern "C" void kernel_launch(void* const* d_in, const int* in_sizes, int n_in,
                              void* d_out, int out_size, void* d_ws, size_t ws_size,
                              hipStream_t stream) {
  (void)in_sizes; (void)n_in; (void)out_size; (void)ws_size;
  const float* obs  = (const float*)d_in[0];
  const float* h0   = (const float*)d_in[1];
  const float* c0   = (const float*)d_in[2];
  const float* Wobs = (const float*)d_in[3];
  const float* bobs = (const float*)d_in[4];
  const float* Wih  = (const float*)d_in[5];
  const float* Whh  = (const float*)d_in[6];
  const float* bih  = (const float*)d_in[7];
  const float* bhh  = (const float*)d_in[8];
  const float* W1   = (const float*)d_in[9];
  const float* ai1  = (const float*)d_in[10];
  const float* aj1  = (const float*)d_in[11];
  const float* b1   = (const float*)d_in[12];
  const float* W2   = (const float*)d_in[13];
  const float* ai2  = (const float*)d_in[14];
  const float* aj2  = (const float*)d_in[15];
  const float* b2   = (const float*)d_in[16];
  const float* Wv   = (const float*)d_in[17];
  const float* bv   = (const float*)d_in[18];
  const float* Wa   = (const float*)d_in[19];
  const float* ba   = (const float*)d_in[20];

  float* out = (float*)d_out;
  // output layout: logp[4096*64], value[4096], h[4096*256], c[4096*256], loss[1]
  float* out_logp  = out;
  float* out_value = out + NAG * ACT;
  float* out_h     = out_value + NAG;
  float* out_c     = out_h + NAG * HID;
  float* out_loss  = out_c + NAG * HID;

  // workspace layout (floats)
  float* W = (float*)d_ws;
  float* ws_enc    = W;                          // 4096*256
  float* ws_gates  = ws_enc    + NAG * HID;      // 4096*1024
  float* ws_comm0  = ws_gates  + NAG * 4 * HID;  // 4096*256  (h copy, quant src 1)
  float* ws_commq  = ws_comm0  + NAG * HID;      // 4096*256  (quantized -> GAT1 in)
  float* ws_hW     = ws_commq  + NAG * HID;      // 4096*256  (hW1, then hW2)
  float* ws_gat1   = ws_hW     + NAG * HID;      // 4096*256  (elu(att+b1), quant src 2)
  float* ws_comm2q = ws_gat1   + NAG * HID;      // 4096*256  (quantized -> GAT2 in)
  float* ws_hc     = ws_comm2q + NAG * HID;      // 4096*512  (concat(h, comm_final))
  float* ws_logits = ws_hc     + NAG * 512;      // 4096*64
  float* ws_ci     = ws_logits + NAG * ACT;      // 4*4096
  float* ws_cj     = ws_ci     + H1 * NAG;       // 4*4096
  float* ws_cjmax  = ws_cj     + H1 * NAG;       // 8
  float* ws_pmin   = ws_cjmax  + 8;              // 1024
  float* ws_pmax   = ws_pmin   + 1024;           // 1024
  float* ws_scale  = ws_pmax   + 1024;           // 4 (scale, zp)
  float* ws_lossA  = ws_scale  + 4;              // 1024
  float* ws_lossB  = ws_lossA  + 1024;           // 1024

  const int LEN = NAG * HID;  // 1,048,576
  const dim3 wave(32, 1, 1);

  // 1) enc = obs @ Wobs + bobs
  gemm16_f32<false, true, false><<<dim3(HID / 16, NAG / 16), wave, 0, stream>>>(
      obs, Wobs, bobs, ws_enc, OBS, OBS, HID, HID);

  // 2) gates = enc @ Wih^T + bih ; gates += h0 @ Whh^T + bhh
  gemm16_f32<true, true, false><<<dim3(4 * HID / 16, NAG / 16), wave, 0, stream>>>(
      ws_enc, Wih, bih, ws_gates, HID, HID, HID, 4 * HID);
  gemm16_f32<true, true, true><<<dim3(4 * HID / 16, NAG / 16), wave, 0, stream>>>(
      h0, Whh, bhh, ws_gates, HID, HID, HID, 4 * HID);

  // 3) LSTM pointwise -> h, c (outputs), concat[:, :256], comm0
  lstm_pointwise<<<dim3(LEN / 256), dim3(256), 0, stream>>>(
      ws_gates, c0, out_h, out_c, ws_hc, ws_comm0);

  // 4) fake-quant round 1 (global min/max of h) + qloss partials A
  minmax_stage1<<<dim3(1024), dim3(256), 0, stream>>>(ws_comm0, ws_pmin, ws_pmax, LEN);
  minmax_stage2<<<dim3(1), dim3(256), 0, stream>>>(ws_pmin, ws_pmax, ws_scale, 1024);
  quant_apply<<<dim3(1024), dim3(256), 0, stream>>>(ws_comm0, ws_commq, ws_scale, ws_lossA, LEN);

  // 5) GAT layer 1
  gemm16_f32<false, false, false><<<dim3(HID / 16, NAG / 16), wave, 0, stream>>>(
      ws_commq, W1, nullptr, ws_hW, HID, HID, HID, HID);
  cicj_kernel<<<dim3((H1 * NAG + 255) / 256), dim3(256), 0, stream>>>(
      ws_hW, ai1, aj1, ws_ci, ws_cj, NAG, H1, GDIM1);
  colmax_kernel<<<dim3(H1), dim3(256), 0, stream>>>(ws_cj, ws_cjmax, NAG);
  gat_attn<1><<<dim3(NAG / 16, H1, 1), wave, 0, stream>>>(
      ws_hW, ws_ci, ws_cj, ws_cjmax, b1, ws_gat1, NAG, GDIM1, HID, HID);

  // 6) fake-quant round 2 + qloss partials B
  minmax_stage1<<<dim3(1024), dim3(256), 0, stream>>>(ws_gat1, ws_pmin, ws_pmax, LEN);
  minmax_stage2<<<dim3(1), dim3(256), 0, stream>>>(ws_pmin, ws_pmax, ws_scale, 1024);
  quant_apply<<<dim3(1024), dim3(256), 0, stream>>>(ws_gat1, ws_comm2q, ws_scale, ws_lossB, LEN);

  // 7) GAT layer 2 (H=1, G=256) -> writes concat[:, 256:512]
  gemm16_f32<false, false, false><<<dim3(HID / 16, NAG / 16), wave, 0, stream>>>(
      ws_comm2q, W2, nullptr, ws_hW, HID, HID, HID, HID);
  cicj_kernel<<<dim3((NAG + 255) / 256), dim3(256), 0, stream>>>(
      ws_hW, ai2, aj2, ws_ci, ws_cj, NAG, 1, HID);
  colmax_kernel<<<dim3(1), dim3(256), 0, stream>>>(ws_cj, ws_cjmax, NAG);
  gat_attn<0><<<dim3(NAG / 16, 1, 4), wave, 0, stream>>>(
      ws_hW, ws_ci, ws_cj, ws_cjmax, b2, ws_hc + HID, NAG, HID, HID, 512);

  // 8) loss = sum(A) + sum(B)
  loss_final<<<dim3(1), dim3(256), 0, stream>>>(ws_lossA, ws_lossB, out_loss, 1024);

  // 9) value head
  value_kernel<<<dim3(NAG / 256), dim3(256), 0, stream>>>(ws_hc, Wv, bv, out_value);

  // 10) logits + log-softmax
  gemm16_f32<false, true, false><<<dim3(ACT / 16, NAG / 16), wave, 0, stream>>>(
      ws_hc, Wa, ba, ws_logits, 512, 512, ACT, ACT);
  logsoftmax_kernel<<<dim3(NAG), dim3(32), 0, stream>>>(ws_logits, out_logp);
}